// InverseMelScale_54391465837000
// MI455X (gfx1250) — compile-verified
//
#include <hip/hip_runtime.h>

// ---------------- problem constants (from reference) ----------------
#define B_SZ   4
#define T_SZ   1024
#define F_SZ   1025      // n_stft
#define M_SZ   128       // n_mels
#define ITERS  20
#define LR     0.3f
#define MOM    0.9f
#define INV    (2.0f / (B_SZ * T_SZ))   // grad prefactor

// ---------------- tiling ----------------
#define ROWS     16      // T-rows per workgroup (= WMMA M)
#define FSTR     1027    // LDS stride for spec/buf   (==3 mod 64: conflict-free)
#define MSTR     131     // LDS stride for mel tile
#define NDSTR    131     // LDS stride for negdiff tile
#define NWAVES   8
#define NTHREADS 256
#define NFT      ((F_SZ + 15) / 16)   // 65 F-tiles in GEMM2

typedef __attribute__((ext_vector_type(2))) float v2f;
typedef __attribute__((ext_vector_type(8))) float v8f;

// D = A(16x4 f32) * B(4x16 f32) + C(16x16 f32)   -> v_wmma_f32_16x16x4_f32
__device__ __forceinline__ v8f wmma_f32(v2f a, v2f b, v8f c) {
  return __builtin_amdgcn_wmma_f32_16x16x4_f32(false, a, false, b,
                                               (short)0, c, false, false);
}

// One-time prologue: fbT[m][f] = fb[f][m]  (coalesced writes; 0.5 MB total)
__global__ __launch_bounds__(NTHREADS)
void transpose_fb_kernel(const float* __restrict__ fb,  // (F, M)
                         float* __restrict__ fbT) {     // (M, F)
  int i = blockIdx.x * NTHREADS + threadIdx.x;
  if (i < M_SZ * F_SZ) {
    int m = i / F_SZ, f = i - m * F_SZ;
    fbT[i] = fb[f * M_SZ + m];
  }
}

__global__ __launch_bounds__(NTHREADS)
void invmel_sgd_kernel(const float* __restrict__ mel,    // (B, M, T)
                       const float* __restrict__ spec0,  // (B, T, F)
                       const float* __restrict__ fb,     // (F, M)
                       const float* __restrict__ fbT,    // (M, F) transposed copy
                       float* __restrict__ out)          // (B, F, T)
{
  __shared__ float s_spec[ROWS * FSTR];   // 65.7 KB
  __shared__ float s_buf [ROWS * FSTR];   // 65.7 KB (momentum)
  __shared__ float s_mel [ROWS * MSTR];   //  8.4 KB (mel_t tile, [t][m])
  __shared__ float s_nd  [ROWS * NDSTR];  //  8.4 KB (negdiff = spec@fb - mel)

  const int tid  = threadIdx.x;
  const int lane = tid & 31;
  const int wave = tid >> 5;
  const int nl   = lane & 15;   // A-row / B-col / C-col index
  const int hs   = lane >> 4;   // half-wave select
  const int khi  = hs * 2;      // K offset for A/B fragments (K=0,1 | 2,3)
  const int hi8  = hs * 8;      // C row offset (M=r | r+8)

  const int blk = blockIdx.x;
  const int b   = blk / (T_SZ / ROWS);
  const int t0  = (blk % (T_SZ / ROWS)) * ROWS;

  // ---- stage spec tile, zero momentum ----
  for (int i = tid; i < ROWS * F_SZ; i += NTHREADS) {
    int r = i / F_SZ, c = i - r * F_SZ;
    s_spec[r * FSTR + c] = spec0[(b * T_SZ + t0 + r) * F_SZ + c];
    s_buf [r * FSTR + c] = 0.0f;
  }
  // ---- stage mel tile transposed: s_mel[t][m] = melspec[b][m][t0+t] ----
  for (int i = tid; i < ROWS * M_SZ; i += NTHREADS) {
    int r = i & (ROWS - 1), m = i >> 4;
    s_mel[r * MSTR + m] = mel[(b * M_SZ + m) * T_SZ + t0 + r];
  }
  __syncthreads();

  for (int it = 0; it < ITERS; ++it) {
    // ============ GEMM1: nd = spec(16xF) @ fb(FxM) - mel  ============
    {
      const int n0 = wave * 16;      // this wave's mel-column tile
      v8f c;
      #pragma unroll
      for (int r = 0; r < 8; ++r)
        c[r] = -s_mel[(r + hi8) * MSTR + n0 + nl];

      #pragma unroll 4
      for (int k = 0; k < 1024; k += 4) {
        v2f a, bf;
        a.x  = s_spec[nl * FSTR + k + khi];
        a.y  = s_spec[nl * FSTR + k + khi + 1];
        bf.x = fb[(k + khi)     * M_SZ + n0 + nl];   // coalesced over nl
        bf.y = fb[(k + khi + 1) * M_SZ + n0 + nl];
        c = wmma_f32(a, bf, c);
      }
      { // epilogue: only K = 1024 is valid (F = 1025)
        v2f a, bf;
        a.x  = (hs == 0) ? s_spec[nl * FSTR + 1024]  : 0.0f;
        a.y  = 0.0f;
        bf.x = (hs == 0) ? fb[1024 * M_SZ + n0 + nl] : 0.0f;
        bf.y = 0.0f;
        c = wmma_f32(a, bf, c);
      }
      #pragma unroll
      for (int r = 0; r < 8; ++r)
        s_nd[(r + hi8) * NDSTR + n0 + nl] = c[r];
    }
    __syncthreads();

    // ==== GEMM2: g = INV * nd(16xM) @ fbT(MxF); fused SGD update ====
    // Single branch-free path: invalid lanes of the last tile load a clamped
    // address and accumulate garbage that is never written back, so EXEC
    // stays all-ones through every WMMA.
    for (int ft = wave; ft < NFT; ft += NWAVES) {
      const int f0   = ft * 16;
      const int fcol = f0 + nl;
      const int fcl  = (fcol < F_SZ) ? fcol : (F_SZ - 1);  // clamp loads only
      v8f acc = {0.f, 0.f, 0.f, 0.f, 0.f, 0.f, 0.f, 0.f};

      #pragma unroll 4
      for (int m = 0; m < M_SZ; m += 4) {
        v2f a, bf;
        a.x  = s_nd[nl * NDSTR + m + khi];
        a.y  = s_nd[nl * NDSTR + m + khi + 1];
        bf.x = fbT[(m + khi)     * F_SZ + fcl];   // coalesced over nl
        bf.y = fbT[(m + khi + 1) * F_SZ + fcl];
        acc = wmma_f32(a, bf, acc);
      }
      if (fcol < F_SZ) {            // per-lane guard only on the write-back
        #pragma unroll
        for (int r = 0; r < 8; ++r) {
          const int off = (r + hi8) * FSTR + fcol;
          float nb = MOM * s_buf[off] + INV * acc[r];
          s_buf[off] = nb;
          float ns = s_spec[off] - LR * nb;
          s_spec[off] = ns > 0.0f ? ns : 0.0f;
        }
      }
    }
    __syncthreads();
  }

  // ---- emit output transposed: out[b][f][t0+t] (t-minor => coalesced) ----
  for (int i = tid; i < ROWS * F_SZ; i += NTHREADS) {
    int r = i & (ROWS - 1), f = i >> 4;
    out[(b * F_SZ + f) * T_SZ + t0 + r] = s_spec[r * FSTR + f];
  }
}

extern "C" void kernel_launch(void* const* d_in, const int* in_sizes, int n_in,
                              void* d_out, int out_size, void* d_ws, size_t ws_size,
                              hipStream_t stream) {
  (void)in_sizes; (void)n_in; (void)out_size; (void)ws_size;
  const float* mel   = (const float*)d_in[0];   // (4,128,1024)
  const float* spec0 = (const float*)d_in[1];   // (4,1024,1025)
  const float* fb    = (const float*)d_in[2];   // (1025,128)
  float* out = (float*)d_out;                   // (4,1025,1024)
  float* fbT = (float*)d_ws;                    // scratch: 128*1025*4 = 524800 B

  // Prologue: build fbT once per call (stream-ordered before the main kernel).
  {
    dim3 grid((M_SZ * F_SZ + NTHREADS - 1) / NTHREADS);
    hipLaunchKernelGGL(transpose_fb_kernel, grid, dim3(NTHREADS), 0, stream,
                       fb, fbT);
  }
  // Main fused 20-iteration SGD, one persistent row-tile per workgroup.
  {
    dim3 grid((B_SZ * T_SZ) / ROWS);   // 256 independent row-tiles
    hipLaunchKernelGGL(invmel_sgd_kernel, grid, dim3(NTHREADS), 0, stream,
                       mel, spec0, fb, fbT, out);
  }
}